// VDP_Dropout_27745488732900
// MI455X (gfx1250) — compile-verified
//
#include <hip/hip_runtime.h>
#include <stdint.h>

// VDP dropout: keep = (u >= 0.1); mu_out = mu*keep/0.9; Sigma_out = Sigma*keep/768.
// Pure streaming kernel: 20 B / element, ~0.2 flop/byte -> HBM-bound (~8.3 us at 23.3 TB/s).
// MI455X strategy: b128 non-temporal loads/stores, wave32 256-thread blocks,
// and the CDNA5 async global->LDS path (ASYNCcnt) to stage the mask input.

typedef __attribute__((ext_vector_type(4))) float v4f;

#define DROP_PROP 0.1f
#define LAST_DIM  768.0f

__global__ __launch_bounds__(256) void vdp_dropout_kernel(
    const float* __restrict__ mu,
    const float* __restrict__ sigma,
    const float* __restrict__ u,
    float* __restrict__ mu_out,
    float* __restrict__ sigma_out,
    int n4)
{
    __shared__ v4f su[256];   // 4 KB: one float4 of `u` per thread

    const int i = blockIdx.x * 256 + threadIdx.x;
    const bool act = (i < n4);   // grid divides exactly for this problem; guard for safety

    // --- CDNA5 async copy: each lane pulls its own 16B of `u` into its LDS slot.
    // the vdst VGPR gives the LDS byte address; low 32 bits of the flat shared pointer are
    // the wave-relative LDS byte offset (aperture carries only the high bits).
    if (act) {
        uint32_t lds_addr = (uint32_t)(uintptr_t)(&su[threadIdx.x]);
        uint64_t gaddr    = (uint64_t)(uintptr_t)(u + (size_t)i * 4);
        asm volatile("global_load_async_to_lds_b128 %0, %1, off"
                     :: "v"(lds_addr), "v"(gaddr)
                     : "memory");
    }

    // --- Direct streaming loads of mu / Sigma (non-temporal: zero reuse,
    // footprint ~194 MB > 192 MB L2). Overlaps with the async copy above.
    v4f m = {}, s = {};
    if (act) {
        m = __builtin_nontemporal_load((const v4f*)mu    + i);
        s = __builtin_nontemporal_load((const v4f*)sigma + i);
    }

    // Wait for this wave's async global->LDS transfers (per-lane slots are
    // wave-private, so no workgroup barrier is required).
    asm volatile("s_wait_asynccnt 0" ::: "memory");

    if (act) {
        v4f uu = su[threadIdx.x];   // ds_load_b128

        v4f om, os;
#pragma unroll
        for (int k = 0; k < 4; ++k) {
            const bool keep = (uu[k] >= DROP_PROP);
            const float mk = keep ? m[k] : 0.0f;   // exact mask multiply
            const float sk = keep ? s[k] : 0.0f;
            // Real fp32 divides to match the reference's rounding exactly
            // (reference: (mu*keep)/0.9 and (Sigma*keep)/768 in float32).
            om[k] = mk / (1.0f - DROP_PROP);
            os[k] = sk / LAST_DIM;
        }

        __builtin_nontemporal_store(om, (v4f*)mu_out    + i);
        __builtin_nontemporal_store(os, (v4f*)sigma_out + i);
    }
}

extern "C" void kernel_launch(void* const* d_in, const int* in_sizes, int n_in,
                              void* d_out, int out_size, void* d_ws, size_t ws_size,
                              hipStream_t stream) {
    const float* mu    = (const float*)d_in[0];
    const float* sigma = (const float*)d_in[1];
    const float* u     = (const float*)d_in[2];

    const int n  = in_sizes[0];   // B*T*D = 9,682,944 (divisible by 4)
    const int n4 = n / 4;         // 2,420,736 = 9456 * 256 -> exact grid

    float* mu_out    = (float*)d_out;       // outputs concatenated: mu then Sigma
    float* sigma_out = (float*)d_out + n;

    const int threads = 256;                // 8 wave32 waves per block
    const int blocks  = (n4 + threads - 1) / threads;

    vdp_dropout_kernel<<<blocks, threads, 0, stream>>>(
        mu, sigma, u, mu_out, sigma_out, n4);
}